// GNNEncoder_489626271957
// MI455X (gfx1250) — compile-verified
//
#include <hip/hip_runtime.h>

// ---------------------------------------------------------------------------
// GraphConv x2 for MI455X (gfx1250).
//   layer: agg = scatter_add(edge_w * x[src]) ; out = agg@W_rel + x@W_root + b
// Bandwidth-bound (≈12 GFLOP vs ≈100-200MB of L2-resident traffic), so we use
// the full-precision f32 matrix path: V_WMMA_F32_16X16X4_F32.
// ---------------------------------------------------------------------------

typedef __attribute__((ext_vector_type(2))) float v2f;
typedef __attribute__((ext_vector_type(8))) float v8f;

#define N_NODES 50000
#define IN_CH   96
#define HID_CH  256
#define OUT_CH  128

__global__ void zero_f32(float* __restrict__ p, int n) {
  int i = blockIdx.x * blockDim.x + threadIdx.x;
  if (i < n) p[i] = 0.0f;
}

// One thread per (edge, 4-channel group): float4 gather from src row (L2-hit),
// 4 non-returning global_atomic_add_f32 into agg[dst].
__global__ void scatter_edges(const float* __restrict__ x,
                              const int* __restrict__ src,
                              const int* __restrict__ dst,
                              const float* __restrict__ ew,
                              float* __restrict__ agg,
                              int nE, int C) {
  long long gid = (long long)blockIdx.x * blockDim.x + threadIdx.x;
  int groups = C >> 2;                   // 4 channels per thread
  long long e = gid / groups;
  if (e >= nE) return;
  int g = (int)(gid - e * groups) << 2;
  int s = src[e];
  int d = dst[e];
  float w = ew[e];
  const float4 v = *reinterpret_cast<const float4*>(x + (size_t)s * C + g);
  float* o = agg + (size_t)d * C + g;
  atomicAdd(o + 0, v.x * w);
  atomicAdd(o + 1, v.y * w);
  atomicAdd(o + 2, v.z * w);
  atomicAdd(o + 3, v.w * w);
}

// Out[16 rows x 64 cols per wave] = A1@W1 + A2@W2 + bias  (optional ReLU).
// WMMA f32 16x16x4 layouts:
//   A (16x4):  lane m = lane&15; VGPR pair = A[m][k0+2*half .. +1]   (contig)
//   B (4x16):  lane n = lane&15; pair = W[k0+2*half][n], W[k0+2*half+1][n]
//   C (16x16): lane n = lane&15; c[r] = C[r + 8*half][n]
__global__ __launch_bounds__(32) void gemm_dual(
    const float* __restrict__ A1, const float* __restrict__ W1,
    const float* __restrict__ A2, const float* __restrict__ W2,
    const float* __restrict__ bias, float* __restrict__ Out,
    int K, int Nout, int relu) {
  const int lane = threadIdx.x;   // wave32, EXEC all-ones
  const int half = lane >> 4;     // 0 or 1
  const int n    = lane & 15;
  const int row0 = blockIdx.x * 16;
  const int col0 = blockIdx.y * 64;

  v8f acc[4];
#pragma unroll
  for (int j = 0; j < 4; ++j) {
    float bv = bias[col0 + j * 16 + n];   // bias depends only on column
#pragma unroll
    for (int r = 0; r < 8; ++r) acc[j][r] = bv;
  }

  const float* a1p = A1 + (size_t)(row0 + n) * K;   // A-frag row = lane&15
  const float* a2p = A2 + (size_t)(row0 + n) * K;

  for (int k0 = 0; k0 < K; k0 += 4) {
    const int ka = k0 + 2 * half;
    v2f a1 = *reinterpret_cast<const v2f*>(a1p + ka);
    v2f a2 = *reinterpret_cast<const v2f*>(a2p + ka);
#pragma unroll
    for (int j = 0; j < 4; ++j) {
      const int c = col0 + j * 16 + n;
      v2f b1, b2;
      b1.x = W1[(size_t)ka * Nout + c];
      b1.y = W1[(size_t)(ka + 1) * Nout + c];
      b2.x = W2[(size_t)ka * Nout + c];
      b2.y = W2[(size_t)(ka + 1) * Nout + c];
      acc[j] = __builtin_amdgcn_wmma_f32_16x16x4_f32(
          false, a1, false, b1, (short)0, acc[j], false, false);
      acc[j] = __builtin_amdgcn_wmma_f32_16x16x4_f32(
          false, a2, false, b2, (short)0, acc[j], false, false);
    }
  }

#pragma unroll
  for (int j = 0; j < 4; ++j) {
#pragma unroll
    for (int r = 0; r < 8; ++r) {
      int row = row0 + r + 8 * half;
      float v = acc[j][r];
      if (relu) v = fmaxf(v, 0.0f);
      Out[(size_t)row * Nout + col0 + j * 16 + n] = v;
    }
  }
}

extern "C" void kernel_launch(void* const* d_in, const int* in_sizes, int n_in,
                              void* d_out, int out_size, void* d_ws, size_t ws_size,
                              hipStream_t stream) {
  const float* x       = (const float*)d_in[0];
  const int*   ei      = (const int*)d_in[1];   // int32 (JAX default x64-off)
  const float* ew      = (const float*)d_in[2];
  const float* W1_rel  = (const float*)d_in[3];
  const float* b1      = (const float*)d_in[4];
  const float* W1_root = (const float*)d_in[5];
  const float* W2_rel  = (const float*)d_in[6];
  const float* b2      = (const float*)d_in[7];
  const float* W2_root = (const float*)d_in[8];
  float* out = (float*)d_out;

  const int nE = in_sizes[1] / 2;
  const int* src = ei;        // edge_index[0]
  const int* dst = ei + nE;   // edge_index[1]

  // Workspace: h [50000x256] then a reusable agg buffer [50000x256]
  // (used first as [50000x96], then zeroed and reused as [50000x256]).
  // Peak: 2 * 50000 * 256 * 4 = 102.4 MB.
  float* h   = (float*)d_ws;
  float* agg = h + (size_t)N_NODES * HID_CH;

  // ------------------------- Layer 1 -------------------------
  {
    int nz = N_NODES * IN_CH;
    zero_f32<<<(nz + 255) / 256, 256, 0, stream>>>(agg, nz);
    long long tot = (long long)nE * (IN_CH / 4);
    scatter_edges<<<(unsigned)((tot + 255) / 256), 256, 0, stream>>>(
        x, src, dst, ew, agg, nE, IN_CH);
    dim3 g(N_NODES / 16, HID_CH / 64);   // 3125 x 4 waves
    gemm_dual<<<g, 32, 0, stream>>>(agg, W1_rel, x, W1_root, b1, h,
                                    IN_CH, HID_CH, /*relu=*/1);
  }
  // ------------------------- Layer 2 -------------------------
  {
    int nz = N_NODES * HID_CH;
    zero_f32<<<(nz + 255) / 256, 256, 0, stream>>>(agg, nz);
    long long tot = (long long)nE * (HID_CH / 4);
    scatter_edges<<<(unsigned)((tot + 255) / 256), 256, 0, stream>>>(
        h, src, dst, ew, agg, nE, HID_CH);
    dim3 g(N_NODES / 16, OUT_CH / 64);   // 3125 x 2 waves
    gemm_dual<<<g, 32, 0, stream>>>(agg, W2_rel, h, W2_root, b2, out,
                                    HID_CH, OUT_CH, /*relu=*/0);
  }
}